// StructuredStateRecurrence_4715874091316
// MI455X (gfx1250) — compile-verified
//
#include <hip/hip_runtime.h>
#include <hip/hip_bf16.h>
#include <stdint.h>

typedef float v2f __attribute__((ext_vector_type(2)));
typedef float v8f __attribute__((ext_vector_type(8)));

#define B_    4
#define L_    4096
#define D_    2048
#define M_    512
#define CL_   128
#define NC_   (L_ / CL_)   // 32 chunks
#define KSTEP 16           // K rows staged in LDS per pipeline stage

__device__ __forceinline__ float sigm(float z) { return 1.0f / (1.0f + expf(-z)); }

// ---------------------------------------------------------------------------
// gfx1250 async copy: global -> LDS, 16B per lane, tracked by ASYNCcnt.
// ---------------------------------------------------------------------------
__device__ __forceinline__ void async_b128(uint32_t lds_addr, const float* gptr) {
    asm volatile("global_load_async_to_lds_b128 %0, %1, off"
                 :: "v"(lds_addr), "v"((unsigned long long)(uintptr_t)gptr)
                 : "memory");
}
__device__ __forceinline__ void wait_async0() {
    asm volatile("s_wait_asynccnt 0x0" ::: "memory");
}

// Prefetch B tile rows [k0, k0+KSTEP) x cols [colBase, colBase+128) into LDS.
// 256 threads: thread t covers rows (t>>5) and (t>>5)+8, cols (t&31)*4 .. +4.
__device__ __forceinline__ void prefetch_b(const float* __restrict__ Bsrc, int ldb,
                                           int k0, int colBase, uint32_t ldsBase,
                                           int tid) {
    const int r  = tid >> 5;          // 0..7
    const int c4 = (tid & 31) * 4;    // 0,4,...,124
    const float* g0 = Bsrc + (size_t)(k0 + r) * ldb + colBase + c4;
    const float* g1 = g0 + (size_t)8 * ldb;
    const uint32_t l0 = ldsBase + (uint32_t)((r * 128 + c4) * 4);
    const uint32_t l1 = l0 + 8u * 128u * 4u;
    async_b128(l0, g0);
    async_b128(l1, g1);
}

// One KSTEP-deep MMA stage: A from global (row-major, leading dim lda, local
// k offset kloc), B fragments from the staged LDS tile.
__device__ __forceinline__ void mma_stage(const float* __restrict__ A, int lda, int kloc,
                                          const float* __restrict__ bt,
                                          int rowBase, int lid, int kh, int wcol,
                                          v8f acc[2][4]) {
    #pragma unroll
    for (int ks = 0; ks < KSTEP; ks += 4) {
        v2f a[2];
        #pragma unroll
        for (int rt = 0; rt < 2; ++rt)
            a[rt] = *(const v2f*)(A + (size_t)(rowBase + rt * 16 + lid) * lda + kloc + ks + 2 * kh);
        v2f bf[4];
        #pragma unroll
        for (int ct = 0; ct < 4; ++ct) {
            const int bi = (ks + 2 * kh) * 128 + wcol + ct * 16 + lid;
            bf[ct].x = bt[bi];
            bf[ct].y = bt[bi + 128];
        }
        #pragma unroll
        for (int rt = 0; rt < 2; ++rt)
            #pragma unroll
            for (int ct = 0; ct < 4; ++ct)
                acc[rt][ct] = __builtin_amdgcn_wmma_f32_16x16x4_f32(
                    false, a[rt], false, bf[ct], (short)0, acc[rt][ct], false, false);
    }
}

// ---------------------------------------------------------------------------
// GEMM1: V[16384 x 512] = X[16384 x 2048] * Wv[2048 x 512] + bv
// 256 threads (8 waves), block tile 128x128, wave tile 32x64,
// B staged in double-buffered LDS via async copies.
// ---------------------------------------------------------------------------
__global__ __launch_bounds__(256) void gemm_v_kernel(
    const float* __restrict__ X, const float* __restrict__ Wv,
    const float* __restrict__ bv, float* __restrict__ V)
{
    const int tid  = threadIdx.x;
    const int lane = tid & 31;
    const int wave = tid >> 5;
    const int lid  = lane & 15;
    const int kh   = lane >> 4;
    const int rowBase  = blockIdx.y * 128 + (wave >> 1) * 32;
    const int bColBase = blockIdx.x * 128;
    const int wcol     = (wave & 1) * 64;

    __shared__ __align__(16) float bT[2][KSTEP * 128];   // 16B-aligned for async b128
    const float* bT0 = &bT[0][0];
    const float* bT1 = &bT[1][0];
    const uint32_t lds0 = (uint32_t)(uintptr_t)&bT[0][0];
    const uint32_t lds1 = (uint32_t)(uintptr_t)&bT[1][0];

    v8f acc[2][4];
    #pragma unroll
    for (int i = 0; i < 2; ++i)
        #pragma unroll
        for (int j = 0; j < 4; ++j) acc[i][j] = (v8f)0.0f;

    prefetch_b(Wv, M_, 0, bColBase, lds0, tid);

    for (int k0 = 0; k0 < D_; k0 += KSTEP) {
        wait_async0();
        __syncthreads();                       // buffer ready + prev buffer drained
        const int cur = (k0 >> 4) & 1;
        if (k0 + KSTEP < D_)
            prefetch_b(Wv, M_, k0 + KSTEP, bColBase, cur ? lds0 : lds1, tid);
        mma_stage(X, D_, k0, cur ? bT1 : bT0, rowBase, lid, kh, wcol, acc);
    }

    #pragma unroll
    for (int rt = 0; rt < 2; ++rt)
        #pragma unroll
        for (int ct = 0; ct < 4; ++ct) {
            const int col  = bColBase + wcol + ct * 16 + lid;
            const float bb = bv[col];
            #pragma unroll
            for (int r = 0; r < 8; ++r) {
                const int row = rowBase + rt * 16 + r + 8 * kh;
                V[(size_t)row * M_ + col] = acc[rt][ct][r] + bb;
            }
        }
}

// ---------------------------------------------------------------------------
// Scan pass 1: per (b, chunk, m) chunk-local scan end value (s0 = 0).
// ---------------------------------------------------------------------------
__global__ __launch_bounds__(256) void scan_pass1(
    const float* __restrict__ V, const float* __restrict__ tdec,
    float* __restrict__ carry)
{
    const int idx = blockIdx.x * blockDim.x + threadIdx.x;
    const int m = idx % M_;
    const int c = (idx / M_) % NC_;
    const int b = idx / (M_ * NC_);
    const float td = sigm(tdec[m]) * 0.9f + 0.1f;
    const float* p = V + (size_t)(b * L_ + c * CL_) * M_ + m;
    float s = 0.0f;
    for (int t = 0; t < CL_; ++t)
        s = fmaf(s, td, p[(size_t)t * M_]);
    carry[((size_t)b * NC_ + c) * M_ + m] = s;
}

// ---------------------------------------------------------------------------
// Scan combine: per (b, m) fold NC_ chunk carries into chunk-entry states.
// ---------------------------------------------------------------------------
__global__ __launch_bounds__(256) void scan_combine(
    const float* __restrict__ tdec, float* __restrict__ carry)
{
    const int idx = blockIdx.x * blockDim.x + threadIdx.x;
    const int m = idx % M_;
    const int b = idx / M_;
    const float td  = sigm(tdec[m]) * 0.9f + 0.1f;
    const float tdc = powf(td, (float)CL_);
    float s = 0.0f;
    for (int c = 0; c < NC_; ++c) {
        const size_t off = ((size_t)b * NC_ + c) * M_ + m;
        const float a = carry[off];
        carry[off] = s;            // entry state of chunk c
        s = fmaf(s, tdc, a);
    }
}

// ---------------------------------------------------------------------------
// Scan pass 2: rescan chunks from entry state; wv = s + ms*tf in place;
// emit next_memory_state from the last time step.
// ---------------------------------------------------------------------------
__global__ __launch_bounds__(256) void scan_pass2(
    float* __restrict__ V, const float* __restrict__ tdec,
    const float* __restrict__ tfir, const float* __restrict__ ms,
    const float* __restrict__ entry, float* __restrict__ nms)
{
    const int idx = blockIdx.x * blockDim.x + threadIdx.x;
    const int m = idx % M_;
    const int c = (idx / M_) % NC_;
    const int b = idx / (M_ * NC_);
    const float td  = sigm(tdec[m]) * 0.9f + 0.1f;
    const float add = ms[b * M_ + m] * sigm(tfir[m]);
    float s = entry[((size_t)b * NC_ + c) * M_ + m];
    float* p = V + (size_t)(b * L_ + c * CL_) * M_ + m;
    for (int t = 0; t < CL_; ++t) {
        s = fmaf(s, td, p[(size_t)t * M_]);
        p[(size_t)t * M_] = s + add;
    }
    if (c == NC_ - 1)
        nms[b * M_ + m] = s + add;
}

// ---------------------------------------------------------------------------
// GEMM2: OUT[16384 x 2048] = X + concat(X, WVbuf)[16384 x 2560] * Wg + bg
// Same pipeline; A source switches from X to WVbuf at k = D_.
// ---------------------------------------------------------------------------
__global__ __launch_bounds__(256) void gemm_out_kernel(
    const float* __restrict__ X, const float* __restrict__ WVbuf,
    const float* __restrict__ Wg, const float* __restrict__ bg,
    float* __restrict__ OUT)
{
    const int tid  = threadIdx.x;
    const int lane = tid & 31;
    const int wave = tid >> 5;
    const int lid  = lane & 15;
    const int kh   = lane >> 4;
    const int rowBase  = blockIdx.y * 128 + (wave >> 1) * 32;
    const int bColBase = blockIdx.x * 128;
    const int wcol     = (wave & 1) * 64;

    __shared__ __align__(16) float bT[2][KSTEP * 128];   // 16B-aligned for async b128
    const float* bT0 = &bT[0][0];
    const float* bT1 = &bT[1][0];
    const uint32_t lds0 = (uint32_t)(uintptr_t)&bT[0][0];
    const uint32_t lds1 = (uint32_t)(uintptr_t)&bT[1][0];

    v8f acc[2][4];
    #pragma unroll
    for (int i = 0; i < 2; ++i)
        #pragma unroll
        for (int j = 0; j < 4; ++j) acc[i][j] = (v8f)0.0f;

    prefetch_b(Wg, D_, 0, bColBase, lds0, tid);

    for (int k0 = 0; k0 < D_ + M_; k0 += KSTEP) {
        wait_async0();
        __syncthreads();
        const int cur = (k0 >> 4) & 1;
        if (k0 + KSTEP < D_ + M_)
            prefetch_b(Wg, D_, k0 + KSTEP, bColBase, cur ? lds0 : lds1, tid);
        const float* bt = cur ? bT1 : bT0;
        if (k0 < D_)
            mma_stage(X,     D_, k0,      bt, rowBase, lid, kh, wcol, acc);
        else
            mma_stage(WVbuf, M_, k0 - D_, bt, rowBase, lid, kh, wcol, acc);
    }

    #pragma unroll
    for (int rt = 0; rt < 2; ++rt)
        #pragma unroll
        for (int ct = 0; ct < 4; ++ct) {
            const int col  = bColBase + wcol + ct * 16 + lid;
            const float bb = bg[col];
            #pragma unroll
            for (int r = 0; r < 8; ++r) {
                const int row = rowBase + rt * 16 + r + 8 * kh;
                const size_t o = (size_t)row * D_ + col;
                OUT[o] = acc[rt][ct][r] + X[o] + bb;
            }
        }
}

// ---------------------------------------------------------------------------
extern "C" void kernel_launch(void* const* d_in, const int* in_sizes, int n_in,
                              void* d_out, int out_size, void* d_ws, size_t ws_size,
                              hipStream_t stream) {
    const float* x    = (const float*)d_in[0];   // [B,L,D]
    const float* ms   = (const float*)d_in[1];   // [B,M]
    const float* Wv   = (const float*)d_in[2];   // [D,M]
    const float* bv   = (const float*)d_in[3];   // [M]
    const float* tdec = (const float*)d_in[4];   // [M]
    const float* tfir = (const float*)d_in[5];   // [M]
    const float* Wg   = (const float*)d_in[6];   // [D+M,D]
    const float* bg   = (const float*)d_in[7];   // [D]

    float* out = (float*)d_out;                       // [B,L,D]
    float* nms = out + (size_t)B_ * L_ * D_;          // [B,M]

    float* v_buf = (float*)d_ws;                      // [B,L,M]  (32 MB)
    float* carry = v_buf + (size_t)B_ * L_ * M_;      // [B,NC,M] (256 KB)

    // 1) v = x @ Wv + bv
    {
        dim3 grid(M_ / 128, (B_ * L_) / 128);         // (4, 128)
        gemm_v_kernel<<<grid, 256, 0, stream>>>(x, Wv, bv, v_buf);
    }
    // 2) chunk-local scans
    scan_pass1<<<(B_ * NC_ * M_) / 256, 256, 0, stream>>>(v_buf, tdec, carry);
    // 3) fold chunk carries -> chunk entry states
    scan_combine<<<(B_ * M_) / 256, 256, 0, stream>>>(tdec, carry);
    // 4) finalize scan, fuse + ms*tf, emit next_memory_state
    scan_pass2<<<(B_ * NC_ * M_) / 256, 256, 0, stream>>>(v_buf, tdec, tfir, ms, carry, nms);
    // 5) out = x + concat(x, wv) @ Wg + bg
    {
        dim3 grid(D_ / 128, (B_ * L_) / 128);         // (16, 128)
        gemm_out_kernel<<<grid, 256, 0, stream>>>(x, v_buf, Wg, bg, out);
    }
}